// MGLSTM_68728066671026
// MI455X (gfx1250) — compile-verified
//
#include <hip/hip_runtime.h>
#include <hip/hip_bf16.h>
#include <math.h>

typedef float v2f __attribute__((ext_vector_type(2)));
typedef float v8f __attribute__((ext_vector_type(8)));

#define DH 64

// ---------------------------------------------------------------------------
// WMMA fp32 GEMM: C[nRows x 64] = A[nRows x K] @ B[K x 64]  (+= C if accumulate)
// One wave (32 lanes) computes one 16x16 tile using V_WMMA_F32_16X16X4_F32.
// nRows % 16 == 0, K % 4 == 0 (holds: 50000, K in {64,128}).
// ---------------------------------------------------------------------------
__global__ void gemm_wmma_f32(const float* __restrict__ A,
                              const float* __restrict__ B,
                              float* __restrict__ C,
                              int nRows, int K, int accumulate) {
    const int wavesPerBlock = blockDim.x >> 5;
    const int wave = blockIdx.x * wavesPerBlock + (threadIdx.x >> 5);
    const int mTiles = nRows >> 4;
    const int nTiles = DH >> 4;              // 4
    if (wave >= mTiles * nTiles) return;     // whole-wave uniform: EXEC stays full

    const int tm = wave / nTiles;
    const int tn = wave - tm * nTiles;
    const int m0 = tm << 4, n0 = tn << 4;

    const int lane = threadIdx.x & 31;
    const int half = lane >> 4;              // 0: K+{0,1}, 1: K+{2,3}
    const int l15  = lane & 15;

    v8f acc = {0.f, 0.f, 0.f, 0.f, 0.f, 0.f, 0.f, 0.f};
    float* Cb = C + (size_t)(m0 + half * 8) * DH + n0 + l15;
    if (accumulate) {
#pragma unroll
        for (int r = 0; r < 8; ++r) acc[r] = Cb[(size_t)r * DH];
    }

    const float* Arow = A + (size_t)(m0 + l15) * K + half * 2;
    const float* Bcol = B + (size_t)(half * 2) * DH + n0 + l15;

#pragma unroll 4
    for (int k0 = 0; k0 < K; k0 += 4) {
        v2f a, b;
        a.x = Arow[k0];
        a.y = Arow[k0 + 1];
        b.x = Bcol[(size_t)k0 * DH];
        b.y = Bcol[(size_t)k0 * DH + DH];
        acc = __builtin_amdgcn_wmma_f32_16x16x4_f32(
            /*neg_a=*/false, a, /*neg_b=*/false, b,
            /*c_mod=*/(short)0, acc, /*reuse_a=*/false, /*reuse_b=*/false);
    }

#pragma unroll
    for (int r = 0; r < 8; ++r) Cb[(size_t)r * DH] = acc[r];
}

// ---------------------------------------------------------------------------
// Attention score projections: s_src[r] = xw[r,:]·a_src ; s_dst[r] = xw[r,:]·a_dst
// ---------------------------------------------------------------------------
__global__ void rowdot_kernel(const float* __restrict__ xw,
                              const float* __restrict__ a_src,
                              const float* __restrict__ a_dst,
                              float* __restrict__ s_src,
                              float* __restrict__ s_dst, int n) {
    int r = blockIdx.x * blockDim.x + threadIdx.x;
    if (r >= n) return;
    const float4* row = (const float4*)(xw + (size_t)r * DH);
    const float4* as  = (const float4*)a_src;
    const float4* ad  = (const float4*)a_dst;
    float ss = 0.f, sd = 0.f;
#pragma unroll
    for (int j = 0; j < DH / 4; ++j) {
        float4 v = row[j], a = as[j], d = ad[j];
        ss += v.x * a.x + v.y * a.y + v.z * a.z + v.w * a.w;
        sd += v.x * d.x + v.y * d.y + v.z * d.z + v.w * d.w;
    }
    s_src[r] = ss;
    s_dst[r] = sd;
}

// ---------------------------------------------------------------------------
// Edge prep: int32 src/dst with self-loops appended
// ---------------------------------------------------------------------------
__global__ void prep_edges(const int* __restrict__ eidx,
                           int* __restrict__ es, int* __restrict__ ed,
                           int E, int N) {
    int i = blockIdx.x * blockDim.x + threadIdx.x;
    int EN = E + N;
    if (i >= EN) return;
    if (i < E) { es[i] = eidx[i]; ed[i] = eidx[E + i]; }
    else       { es[i] = i - E;   ed[i] = i - E; }
}

__global__ void fill_f32(float* __restrict__ p, float v, int n) {
    int i = blockIdx.x * blockDim.x + threadIdx.x;
    if (i < n) p[i] = v;
}

// acc[i] = b1[i%64] (+ b2[i%64])  -- GAT output bias pre-load
__global__ void init_bias(float* __restrict__ acc, const float* __restrict__ b1,
                          const float* __restrict__ b2, int total) {
    int i = blockIdx.x * blockDim.x + threadIdx.x;
    if (i >= total) return;
    float v = b1[i & (DH - 1)];
    if (b2) v += b2[i & (DH - 1)];
    acc[i] = v;
}

// ---------------------------------------------------------------------------
// Segment softmax over dst. Pass 1: per-dst max. Pass 2: w=exp(e-max) stored
// per edge + per-dst sum. Pass 3: alpha = w/den, 32 lanes/edge float2 gather.
// ---------------------------------------------------------------------------
__device__ __forceinline__ float edge_score(const float* ss, const float* sd,
                                            int s, int d) {
    float e = ss[s] + sd[d];
    return e > 0.f ? e : 0.2f * e;   // leaky_relu(0.2)
}

__device__ __forceinline__ void atomicMaxF(float* addr, float val) {
    if (val >= 0.f) atomicMax((int*)addr, __float_as_int(val));
    else            atomicMin((unsigned int*)addr, __float_as_uint(val));
}

__global__ void edge_max(const int* __restrict__ es, const int* __restrict__ ed,
                         const float* __restrict__ ss, const float* __restrict__ sd,
                         float* __restrict__ mx, int EN) {
    int i = blockIdx.x * blockDim.x + threadIdx.x;
    if (i >= EN) return;
    atomicMaxF(&mx[ed[i]], edge_score(ss, sd, es[i], ed[i]));
}

__global__ void edge_expsum(const int* __restrict__ es, const int* __restrict__ ed,
                            const float* __restrict__ ss, const float* __restrict__ sd,
                            const float* __restrict__ mx, float* __restrict__ den,
                            float* __restrict__ wexp, int EN) {
    int i = blockIdx.x * blockDim.x + threadIdx.x;
    if (i >= EN) return;
    int d = ed[i];
    float w = __expf(edge_score(ss, sd, es[i], d) - mx[d]);
    wexp[i] = w;
    atomicAdd(&den[d], w);
}

// 32 lanes per edge, 2 features per lane: acc[dst,f] += alpha * xw[src,f]
__global__ void edge_aggregate(const int* __restrict__ es, const int* __restrict__ ed,
                               const float* __restrict__ wexp,
                               const float* __restrict__ den,
                               const float* __restrict__ xw, float* __restrict__ acc,
                               int EN) {
    long long gid = (long long)blockIdx.x * blockDim.x + threadIdx.x;
    int i    = (int)(gid >> 5);
    int lane = (int)(gid & 31);
    if (i >= EN) return;
    int s = es[i], d = ed[i];                 // wave-uniform scalar loads
    float alpha = wexp[i] / den[d];
    float2 v = ((const float2*)(xw + (size_t)s * DH))[lane];
    float* drow = acc + (size_t)d * DH + lane * 2;
    atomicAdd(drow,     alpha * v.x);
    atomicAdd(drow + 1, alpha * v.y);
}

// ---------------------------------------------------------------------------
// h_N = h + tanh(t + b1 + b2)   (gamma branch is dead math in the reference)
// ---------------------------------------------------------------------------
__global__ void hN_kernel(const float* __restrict__ h, const float* __restrict__ t,
                          const float* __restrict__ b1, const float* __restrict__ b2,
                          float* __restrict__ hN, int total) {
    int i = blockIdx.x * blockDim.x + threadIdx.x;
    if (i >= total) return;
    hN[i] = h[i] + tanhf(t[i] + b1[i & (DH - 1)] + b2[i & (DH - 1)]);
}

__device__ __forceinline__ float sigmoidf(float x) { return 1.f / (1.f + __expf(-x)); }

__global__ void lstm_combine(const float* __restrict__ gf, const float* __restrict__ gi,
                             const float* __restrict__ gc, const float* __restrict__ go,
                             const float* __restrict__ c, float* __restrict__ out,
                             int total) {
    int i = blockIdx.x * blockDim.x + threadIdx.x;
    if (i >= total) return;
    float f  = sigmoidf(gf[i]);
    float in = sigmoidf(gi[i]);
    float ct = tanhf(gc[i]);
    float o  = sigmoidf(go[i]);
    float cn = f * c[i] + in * ct;
    out[total + i] = cn;             // c_new
    out[i] = o * tanhf(cn);          // h_new
}

// ---------------------------------------------------------------------------
// Host side
// ---------------------------------------------------------------------------
static inline int cdiv(long long a, long long b) { return (int)((a + b - 1) / b); }

struct SoftmaxScratch {
    float *ssrc, *sdst, *mx, *den, *wexp;
    const int *es, *ed;
    int N, EN;
};

static void run_gat_aggregate(const float* xw, const float* a_src, const float* a_dst,
                              float* acc, const SoftmaxScratch& w, hipStream_t st) {
    const int B = 256;
    rowdot_kernel<<<cdiv(w.N, B), B, 0, st>>>(xw, a_src, a_dst, w.ssrc, w.sdst, w.N);
    fill_f32<<<cdiv(w.N, B), B, 0, st>>>(w.mx, -INFINITY, w.N);
    fill_f32<<<cdiv(w.N, B), B, 0, st>>>(w.den, 0.f, w.N);
    edge_max<<<cdiv(w.EN, B), B, 0, st>>>(w.es, w.ed, w.ssrc, w.sdst, w.mx, w.EN);
    edge_expsum<<<cdiv(w.EN, B), B, 0, st>>>(w.es, w.ed, w.ssrc, w.sdst, w.mx,
                                             w.den, w.wexp, w.EN);
    edge_aggregate<<<cdiv((long long)w.EN * 32, B), B, 0, st>>>(
        w.es, w.ed, w.wexp, w.den, xw, acc, w.EN);
}

extern "C" void kernel_launch(void* const* d_in, const int* in_sizes, int n_in,
                              void* d_out, int out_size, void* d_ws, size_t ws_size,
                              hipStream_t stream) {
    const int N   = in_sizes[2] / DH;        // 50000
    const int E   = in_sizes[1] / 2;         // 800000
    const int DIN = in_sizes[0] / N;         // 128
    const int EN  = E + N;
    const int tot = N * DH;
    const int B   = 256;

    const float* x    = (const float*)d_in[0];
    const int*   eidx = (const int*)d_in[1];
    const float* h    = (const float*)d_in[2];
    const float* c    = (const float*)d_in[3];
    // d_in[4..7]   : gat_input {W, a_src, a_dst, b}  (dead code in reference)
    // gat param packs, dict insertion order {W, a_src, a_dst, b}:
    auto gatW = [&](int base) { return (const float*)d_in[base]; };
    auto gatAs = [&](int base) { return (const float*)d_in[base + 1]; };
    auto gatAd = [&](int base) { return (const float*)d_in[base + 2]; };
    auto gatB = [&](int base) { return (const float*)d_in[base + 3]; };
    const int HID = 8;                        // gat_hidden
    const int GX[4] = {12, 20, 36, 28};       // forget_x, input_x, candidate_x, output_x
    const int GH[4] = {16, 24, 40, 32};       // forget_h, input_h, candidate_h, output_h
    // d_in[44..47] : W_gamma1/2 (dead). 48/49: W_beta1 {W,b}. 50/51: W_beta2 {W,b}.
    const float* Wb1 = (const float*)d_in[48];
    const float* bb1 = (const float*)d_in[49];
    const float* Wb2 = (const float*)d_in[50];
    const float* bb2 = (const float*)d_in[51];

    // ---- carve workspace ----
    char* w = (char*)d_ws;
    auto carve = [&](size_t bytes) {
        void* p = (void*)w;
        w += (bytes + 255) & ~(size_t)255;
        return p;
    };
    int*   es   = (int*)carve((size_t)EN * 4);
    int*   ed   = (int*)carve((size_t)EN * 4);
    float* wexp = (float*)carve((size_t)EN * 4);
    float* xwB  = (float*)carve((size_t)tot * 4);   // x@W / t1 scratch
    float* hwB  = (float*)carve((size_t)tot * 4);   // h@W scratch
    float* accH = (float*)carve((size_t)tot * 4);   // h_N0 (hidden GAT out)
    float* hN   = (float*)carve((size_t)tot * 4);
    float* ssrc = (float*)carve((size_t)N * 4);
    float* sdst = (float*)carve((size_t)N * 4);
    float* mx   = (float*)carve((size_t)N * 4);
    float* den  = (float*)carve((size_t)N * 4);
    float* gate[4];
    for (int g = 0; g < 4; ++g) gate[g] = (float*)carve((size_t)tot * 4);

    SoftmaxScratch sw{ssrc, sdst, mx, den, wexp, es, ed, N, EN};

    prep_edges<<<cdiv(EN, B), B, 0, stream>>>(eidx, es, ed, E, N);

    const int gemmTiles = (N / 16) * (DH / 16);
    const int gemmBlocks = cdiv(gemmTiles, B / 32);

    // ---- hidden GAT: h_N0 = GAT(gat_hidden, h) ----
    gemm_wmma_f32<<<gemmBlocks, B, 0, stream>>>(h, gatW(HID), hwB, N, DH, 0);
    init_bias<<<cdiv(tot, B), B, 0, stream>>>(accH, gatB(HID), nullptr, tot);
    run_gat_aggregate(hwB, gatAs(HID), gatAd(HID), accH, sw, stream);

    // ---- h_N = h + tanh(h@Wb1 + h_N0@Wb2 + bb1 + bb2) ----
    gemm_wmma_f32<<<gemmBlocks, B, 0, stream>>>(h, Wb1, xwB, N, DH, 0);
    gemm_wmma_f32<<<gemmBlocks, B, 0, stream>>>(accH, Wb2, xwB, N, DH, 1);
    hN_kernel<<<cdiv(tot, B), B, 0, stream>>>(h, xwB, bb1, bb2, hN, tot);

    // ---- 4 gates: gate[g] = GAT(x-params, x) + GAT(h-params, h_N) (pre-activation) ----
    for (int g = 0; g < 4; ++g) {
        gemm_wmma_f32<<<gemmBlocks, B, 0, stream>>>(x, gatW(GX[g]), xwB, N, DIN, 0);
        gemm_wmma_f32<<<gemmBlocks, B, 0, stream>>>(hN, gatW(GH[g]), hwB, N, DH, 0);
        init_bias<<<cdiv(tot, B), B, 0, stream>>>(gate[g], gatB(GX[g]), gatB(GH[g]), tot);
        run_gat_aggregate(xwB, gatAs(GX[g]), gatAd(GX[g]), gate[g], sw, stream);
        run_gat_aggregate(hwB, gatAs(GH[g]), gatAd(GH[g]), gate[g], sw, stream);
    }

    // ---- LSTM combine: out = [h_new | c_new] ----
    lstm_combine<<<cdiv(tot, B), B, 0, stream>>>(
        gate[0], gate[1], gate[2], gate[3], c, (float*)d_out, tot);

    (void)n_in; (void)out_size; (void)ws_size;
}